// DataProxCGLayer_43671227466002
// MI455X (gfx1250) — compile-verified
//
#include <hip/hip_runtime.h>
#include <math.h>

// ---------------- problem constants (match reference) ----------------
#define BN 4
#define CN 12
#define MN 2
#define HN 320
#define WN 320
#define HW (HN*WN)          // 102400
#define NX (BN*MN*HW)       // 819200 complex elems (x/p/r/q)
#define NCOIL (BN*CN*HW)    // 4915200 complex elems (coil buffer)
#define NIMG (BN*CN)        // 48 images to FFT
#define TOLF 1e-6f
#define MAXIT 10
#define RBLK 128            // partial blocks per batch for reductions
#define FFT_SCALE 0.05590169943749474f   // 1/sqrt(320), per 1D pass (ortho)

typedef __attribute__((ext_vector_type(2))) float v2f;
typedef __attribute__((ext_vector_type(8))) float v8f;

// ---------------- complex helpers ----------------
__device__ __forceinline__ float2 cadd(float2 a, float2 b){ return make_float2(a.x+b.x, a.y+b.y); }
__device__ __forceinline__ float2 csub(float2 a, float2 b){ return make_float2(a.x-b.x, a.y-b.y); }
__device__ __forceinline__ float2 cmul(float2 a, float2 b){ return make_float2(a.x*b.x-a.y*b.y, a.x*b.y+a.y*b.x); }
__device__ __forceinline__ float2 cmulc(float2 a, float2 b){ // a*conj(b)
  return make_float2(a.x*b.x+a.y*b.y, a.y*b.x-a.x*b.y); }
__device__ __forceinline__ float2 cscale(float2 a, float s){ return make_float2(a.x*s, a.y*s); }

__device__ __forceinline__ v8f wmma4(v2f a, v2f b, v8f c) {
  // D(16x16,f32) = A(16x4,f32) * B(4x16,f32) + C
  return __builtin_amdgcn_wmma_f32_16x16x4_f32(false, a, false, b, (short)0, c, false, false);
}

// ---------------- 320-point FFT: 16 (WMMA) x 20 (Stockham 4*5), one wave per line ----------------
// A: input line x[0..319] (overwritten as scratch); B: output (natural order)
// tw[r] = cis(+2*pi*r/320); sign = -1 forward, +1 inverse (applied to tw.y at use).
__device__ __forceinline__ void fft320(float2 (&A)[320], float2 (&B)[320],
                                       const float2 (&tw)[320], int lane, float sign) {
  const int mRow  = lane & 15;
  const int hHalf = lane >> 4;

  // F16 matrix fragments: F[m][k] = cis(sign*2*pi*m*k/16), k = 4c + 2h + v (ISA f32 A 16x4 layout)
  v2f Ar[4], Ai[4];
  #pragma unroll
  for (int c = 0; c < 4; ++c) {
    #pragma unroll
    for (int v = 0; v < 2; ++v) {
      int k = 4*c + 2*hHalf + v;
      float2 t = tw[((mRow * k) & 15) * 20];
      Ar[c][v] = t.x; Ai[c][v] = sign * t.y;
    }
  }

  // Stage 1: Y[k1][n2] = sum_n1 F[k1][n1] * x[20*n1 + n2]; two N-tiles (n2 0..15, 16..19 padded)
  #pragma unroll
  for (int t = 0; t < 2; ++t) {
    v8f Yr, Yi;
    #pragma unroll
    for (int j = 0; j < 8; ++j) { Yr[j] = 0.f; Yi[j] = 0.f; }
    #pragma unroll
    for (int c = 0; c < 4; ++c) {
      v2f Br, Bi, nBi;
      #pragma unroll
      for (int v = 0; v < 2; ++v) {
        int k = 4*c + 2*hHalf + v;          // B 4x16 f32 layout: n = lane&15, k = 2h+v per VGPR
        int idx = 20*k + 16*t + mRow;
        idx = idx > 319 ? 319 : idx;        // pad columns (discarded at store)
        float2 xv = A[idx];
        Br[v] = xv.x; Bi[v] = xv.y; nBi[v] = -xv.y;
      }
      Yr = wmma4(Ar[c], Br,  Yr);   // + Fr*Xr
      Yr = wmma4(Ai[c], nBi, Yr);   // - Fi*Xi
      Yi = wmma4(Ar[c], Bi,  Yi);   // + Fr*Xi
      Yi = wmma4(Ai[c], Br,  Yi);   // + Fi*Xr
    }
    // Store with W320^{k1*n2} twiddle: D layout n = lane&15, m = j + 8h
    int n2 = 16*t + mRow;
    if (n2 < 20) {
      #pragma unroll
      for (int j = 0; j < 8; ++j) {
        int k1 = j + 8*hHalf;
        float2 w = tw[(k1 * n2) % 320];
        B[k1*20 + n2] = cmul(make_float2(Yr[j], Yi[j]), make_float2(w.x, sign*w.y));
      }
    }
  }

  // Stage 2a: length-20 Stockham radix-4 (Ns=1): per k1, 5 butterflies, stride 5
  for (int task = lane; task < 80; task += 32) {
    int k1 = task & 15, j = task >> 4;     // j in [0,5)
    float2 a = B[k1*20 + j];
    float2 b = B[k1*20 + j + 5];
    float2 c = B[k1*20 + j + 10];
    float2 d = B[k1*20 + j + 15];
    float2 t0 = cadd(a,c), t1 = csub(a,c), t2 = cadd(b,d), t3 = csub(b,d);
    float2 it3 = make_float2(-sign*t3.y, sign*t3.x);   // sign*i*t3
    A[k1*20 + 4*j + 0] = cadd(t0,t2);
    A[k1*20 + 4*j + 1] = cadd(t1,it3);
    A[k1*20 + 4*j + 2] = csub(t0,t2);
    A[k1*20 + 4*j + 3] = csub(t1,it3);
  }

  // Stage 2b: radix-5 (Ns=4) with twiddles; scatter straight to natural order out[k1 + 16*k2]
  const float c5[5] = {1.f, 0.30901699f, -0.80901699f, -0.80901699f, 0.30901699f};
  const float s5[5] = {0.f, 0.95105652f,  0.58778525f, -0.58778525f, -0.95105652f};
  for (int task = lane; task < 64; task += 32) {
    int k1 = task & 15, j = task >> 4;     // j in [0,4)
    float2 u[5];
    #pragma unroll
    for (int i = 0; i < 5; ++i) {
      u[i] = A[k1*20 + j + 4*i];
      if (i) {
        float2 w = tw[((j*i) % 20) * 16];
        u[i] = cmul(u[i], make_float2(w.x, sign*w.y));
      }
    }
    #pragma unroll
    for (int k = 0; k < 5; ++k) {
      float2 acc = u[0];
      #pragma unroll
      for (int n = 1; n < 5; ++n) {
        int rm = (n*k) % 5;
        acc = cadd(acc, cmul(u[n], make_float2(c5[rm], sign*s5[rm])));
      }
      B[k1 + 16*(j + 4*k)] = acc;
    }
  }
}

// ---------------- batched line-FFT kernel (rows: COL=0, cols: COL=1) ----------------
// 8 lines per workgroup (1 wave each), in-place on `data`. Optional mask fused into store.
template<int COL>
__global__ __launch_bounds__(256)
void k_fft(float2* data, float sign, float scale, const float* mask, const int* gate) {
  if (gate && !*gate) return;
  const int GRP = WN / 8;                       // 40 line-groups per image
  int img = blockIdx.x / GRP;
  int grp = blockIdx.x % GRP;
  float2* base = data + (size_t)img * HW;
  const float* mbase = mask ? (mask + (size_t)img * HW) : nullptr;
  __shared__ float2 bufA[8][320];
  __shared__ float2 bufB[8][320];
  __shared__ float2 twsh[320];

  const float PI2 = 6.283185307179586f;
  for (int i = threadIdx.x; i < 320; i += 256) {
    float s, c; sincosf(PI2 * (float)i * (1.0f/320.0f), &s, &c);
    twsh[i] = make_float2(c, s);
  }
  for (int idx = threadIdx.x; idx < 8*320; idx += 256) {
    int line, elem, g;
    if (COL) { line = idx & 7; elem = idx >> 3; g = elem*WN + grp*8 + line; }  // coalesced 64B
    else     { elem = idx % 320; line = idx / 320; g = (grp*8 + line)*WN + elem; }
    bufA[line][elem] = base[g];
  }
  __syncthreads();

  fft320(bufA[threadIdx.x >> 5], bufB[threadIdx.x >> 5], twsh, threadIdx.x & 31, sign);
  __syncthreads();

  for (int idx = threadIdx.x; idx < 8*320; idx += 256) {
    int line, elem, g;
    if (COL) { line = idx & 7; elem = idx >> 3; g = elem*WN + grp*8 + line; }
    else     { elem = idx % 320; line = idx / 320; g = (grp*8 + line)*WN + elem; }
    float2 v = cscale(bufB[line][elem], scale);
    if (mbase) v = cscale(v, mbase[g]);
    base[g] = v;
  }
}

// ---------------- coil expand: coil[b,c] = sum_m p[b,m] * sm[b,c,m] ----------------
__global__ __launch_bounds__(256)
void k_coils(const float2* __restrict__ p, const float2* __restrict__ sm,
             float2* __restrict__ coil, const int* gate) {
  if (gate && !*gate) return;
  int n = blockIdx.x*blockDim.x + threadIdx.x;
  if (n >= NCOIL) return;
  int bc = n / HW, pos = n % HW;
  int b = bc / CN;
  __builtin_prefetch(&sm[(size_t)(bc*MN)*HW + pos + 4096], 0, 1);
  float2 acc = make_float2(0.f, 0.f);
  #pragma unroll
  for (int m = 0; m < MN; ++m) {
    float2 pv = p[(size_t)(b*MN + m)*HW + pos];
    float2 sv = sm[(size_t)(bc*MN + m)*HW + pos];
    acc = cadd(acc, cmul(pv, sv));
  }
  coil[n] = acc;
}

// ---------------- coil combine: dst = lam * sum_c conj(sm)*coil + add ----------------
__global__ __launch_bounds__(256)
void k_at(const float2* __restrict__ coil, const float2* __restrict__ sm,
          const float* __restrict__ lam, const float2* __restrict__ add,
          float2* dst1, float2* dst2, const int* gate) {
  if (gate && !*gate) return;
  int n = blockIdx.x*blockDim.x + threadIdx.x;
  if (n >= NX) return;
  int bm = n / HW, pos = n % HW;
  int b = bm / MN, m = bm % MN;
  float2 acc = make_float2(0.f, 0.f);
  #pragma unroll
  for (int c = 0; c < CN; ++c) {
    int bc = b*CN + c;
    float2 cv = coil[(size_t)bc*HW + pos];
    float2 sv = sm[(size_t)(bc*MN + m)*HW + pos];
    acc = cadd(acc, cmulc(cv, sv));
  }
  float L = *lam;
  float2 o = cadd(cscale(acc, L), add[n]);
  dst1[n] = o;
  if (dst2) dst2[n] = o;
}

// ---------------- small elementwise kernels ----------------
__global__ void k_zero(float* o, int n){ int i = blockIdx.x*blockDim.x + threadIdx.x; if (i<n) o[i]=0.f; }

__global__ void k_ymask(const float2* __restrict__ y, const float* __restrict__ mk, float2* coil){
  int n = blockIdx.x*blockDim.x + threadIdx.x;
  if (n < NCOIL) coil[n] = cscale(y[n], mk[n]);
}

__global__ void k_axpy(float2* x, float2* r, const float2* __restrict__ p, const float2* __restrict__ q,
                       const float2* __restrict__ alpha, const int* gate){
  if (!*gate) return;
  int n = blockIdx.x*blockDim.x + threadIdx.x;
  if (n >= NX) return;
  float2 a = alpha[n / (MN*HW)];
  x[n] = cadd(x[n], cmul(a, p[n]));
  r[n] = csub(r[n], cmul(a, q[n]));
}

__global__ void k_pup(float2* p, const float2* __restrict__ r, const float* __restrict__ beta, const int* gate){
  if (!*gate) return;
  int n = blockIdx.x*blockDim.x + threadIdx.x;
  if (n >= NX) return;
  p[n] = cadd(r[n], cscale(p[n], beta[n / (MN*HW)]));
}

// ---------------- deterministic two-stage reductions ----------------
__global__ __launch_bounds__(256)
void k_norm2(const float2* __restrict__ v, float* part, const int* gate){
  if (gate && !*gate) return;
  int b = blockIdx.y;
  const int perB = MN*HW, chunk = perB / RBLK;        // 204800 / 128 = 1600
  int s0 = blockIdx.x*chunk, s1 = s0 + chunk;
  const float2* vb = v + (size_t)b*perB;
  float acc = 0.f;
  for (int i = s0 + threadIdx.x; i < s1; i += 256) { float2 t = vb[i]; acc += t.x*t.x + t.y*t.y; }
  __shared__ float sh[256];
  sh[threadIdx.x] = acc; __syncthreads();
  for (int s = 128; s > 0; s >>= 1) { if (threadIdx.x < s) sh[threadIdx.x] += sh[threadIdx.x+s]; __syncthreads(); }
  if (!threadIdx.x) part[b*RBLK + blockIdx.x] = sh[0];
}

__global__ __launch_bounds__(256)
void k_dotc(const float2* __restrict__ p, const float2* __restrict__ q, float2* part, const int* gate){
  if (!*gate) return;
  int b = blockIdx.y;
  const int perB = MN*HW, chunk = perB / RBLK;
  int s0 = blockIdx.x*chunk, s1 = s0 + chunk;
  const float2* pb = p + (size_t)b*perB;
  const float2* qb = q + (size_t)b*perB;
  float ar = 0.f, ai = 0.f;
  for (int i = s0 + threadIdx.x; i < s1; i += 256) {
    float2 pv = pb[i], qv = qb[i];
    ar += pv.x*qv.x + pv.y*qv.y;     // Re(p*conj(q))
    ai += pv.y*qv.x - pv.x*qv.y;     // Im(p*conj(q))
  }
  __shared__ float shr[256], shi[256];
  shr[threadIdx.x] = ar; shi[threadIdx.x] = ai; __syncthreads();
  for (int s = 128; s > 0; s >>= 1) {
    if (threadIdx.x < s) { shr[threadIdx.x] += shr[threadIdx.x+s]; shi[threadIdx.x] += shi[threadIdx.x+s]; }
    __syncthreads();
  }
  if (!threadIdx.x) part[b*RBLK + blockIdx.x] = make_float2(shr[0], shi[0]);
}

__global__ void k_fin1(const float* __restrict__ part, float* out, float* out2, const int* gate){
  if (gate && !*gate) return;
  int b = blockIdx.x;
  __shared__ float sh[RBLK];
  sh[threadIdx.x] = part[b*RBLK + threadIdx.x]; __syncthreads();
  for (int s = RBLK/2; s > 0; s >>= 1) { if (threadIdx.x < s) sh[threadIdx.x] += sh[threadIdx.x+s]; __syncthreads(); }
  if (!threadIdx.x) { out[b] = sh[0]; if (out2) out2[b] = sh[0]; }
}

__global__ void k_fin2(const float2* __restrict__ part, float2* out, const int* gate){
  if (!*gate) return;
  int b = blockIdx.x;
  __shared__ float shr[RBLK], shi[RBLK];
  float2 t = part[b*RBLK + threadIdx.x];
  shr[threadIdx.x] = t.x; shi[threadIdx.x] = t.y; __syncthreads();
  for (int s = RBLK/2; s > 0; s >>= 1) {
    if (threadIdx.x < s) { shr[threadIdx.x] += shr[threadIdx.x+s]; shi[threadIdx.x] += shi[threadIdx.x+s]; }
    __syncthreads();
  }
  if (!threadIdx.x) out[b] = make_float2(shr[0], shi[0]);
}

// ---------------- CG scalar kernels ----------------
__global__ void k_flag(const float* rr, const float* x0x0, int* flag){
  float mn = 3.4e38f;
  for (int b = 0; b < BN; ++b) mn = fminf(mn, rr[b] / x0x0[b]);
  *flag = (mn > TOLF) ? 1 : 0;
}

__global__ void k_alpha(const float* rr, const float2* pq, float2* alpha, const int* gate){
  if (!*gate) return;
  int b = threadIdx.x;
  if (b < BN) {
    float2 d = pq[b];
    float den = d.x*d.x + d.y*d.y;
    float rv = rr[b];
    alpha[b] = make_float2(rv*d.x/den, -rv*d.y/den);   // rr * conj(pq) / |pq|^2
  }
}

__global__ void k_beta(float* rr, const float* rrn, float* beta, const int* gate){
  if (!*gate) return;
  for (int b = 0; b < BN; ++b) { beta[b] = rrn[b] / rr[b]; rr[b] = rrn[b]; }
}

// ---------------- host orchestration ----------------
extern "C" void kernel_launch(void* const* d_in, const int* in_sizes, int n_in,
                              void* d_out, int out_size, void* d_ws, size_t ws_size,
                              hipStream_t stream) {
  const float2* zin = (const float2*)d_in[0];   // x  (B,M,H,W,2)
  const float2* y   = (const float2*)d_in[1];   // y  (B,C,H,W,2)
  const float2* sm  = (const float2*)d_in[2];   // smaps (B,C,M,H,W,2)
  const float*  mk  = (const float*)d_in[3];    // mask (B,C,H,W,1)
  const float*  lam = (const float*)d_in[4];    // lambda_a (1,)
  float2* x = (float2*)d_out;                   // solution lives in d_out

  char* ws = (char*)d_ws;
  size_t off = 0;
  auto alloc = [&](size_t bytes) -> void* {
    off = (off + 255) & ~(size_t)255;
    void* ptr = ws + off;
    off += bytes;
    return ptr;
  };
  float2* coil  = (float2*)alloc((size_t)NCOIL * 8);
  float2* p     = (float2*)alloc((size_t)NX * 8);
  float2* r     = (float2*)alloc((size_t)NX * 8);
  float2* q     = (float2*)alloc((size_t)NX * 8);
  float*  partN = (float*)alloc((size_t)BN * RBLK * 4);
  float2* partC = (float2*)alloc((size_t)BN * RBLK * 8);
  float*  x0x0  = (float*)alloc(BN * 4);
  float*  rr    = (float*)alloc(BN * 4);
  float*  rrn   = (float*)alloc(BN * 4);
  float*  betaB = (float*)alloc(BN * 4);
  float2* pq    = (float2*)alloc(BN * 8);
  float2* alphaB= (float2*)alloc(BN * 8);
  int*    flag  = (int*)alloc(sizeof(int));
  (void)ws_size; (void)in_sizes; (void)n_in; (void)out_size;

  const int FFTG = NIMG * (WN/8);                       // 1920 blocks per FFT pass
  auto EW = [](int n){ return (n + 255) / 256; };
  const dim3 RED(RBLK, BN);

  // ---- setup: x = 0; x0 = lam*AT(y) + z; r = p = x0; rr = x0x0 = ||x0||^2 ----
  k_zero <<<EW(2*NX), 256, 0, stream>>>((float*)x, 2*NX);
  k_ymask<<<EW(NCOIL), 256, 0, stream>>>(y, mk, coil);
  k_fft<0><<<FFTG, 256, 0, stream>>>(coil, +1.f, FFT_SCALE, nullptr, nullptr);  // ifft rows
  k_fft<1><<<FFTG, 256, 0, stream>>>(coil, +1.f, FFT_SCALE, nullptr, nullptr);  // ifft cols
  k_at   <<<EW(NX), 256, 0, stream>>>(coil, sm, lam, zin, p, r, nullptr);
  k_norm2<<<RED, 256, 0, stream>>>(r, partN, nullptr);
  k_fin1 <<<BN, RBLK, 0, stream>>>(partN, x0x0, rr, nullptr);

  // ---- CG iterations ----
  for (int it = 0; it < MAXIT; ++it) {
    k_flag <<<1, 1, 0, stream>>>(rr, x0x0, flag);
    // q = lam * AT(A(p)) + p
    k_coils<<<EW(NCOIL), 256, 0, stream>>>(p, sm, coil, flag);
    k_fft<0><<<FFTG, 256, 0, stream>>>(coil, -1.f, FFT_SCALE, nullptr, flag);   // fft rows
    k_fft<1><<<FFTG, 256, 0, stream>>>(coil, -1.f, FFT_SCALE, mk, flag);        // fft cols + mask
    k_fft<1><<<FFTG, 256, 0, stream>>>(coil, +1.f, FFT_SCALE, nullptr, flag);   // ifft cols
    k_fft<0><<<FFTG, 256, 0, stream>>>(coil, +1.f, FFT_SCALE, nullptr, flag);   // ifft rows
    k_at   <<<EW(NX), 256, 0, stream>>>(coil, sm, lam, p, q, nullptr, flag);
    // alpha = rr / <p, q>
    k_dotc <<<RED, 256, 0, stream>>>(p, q, partC, flag);
    k_fin2 <<<BN, RBLK, 0, stream>>>(partC, pq, flag);
    k_alpha<<<1, BN, 0, stream>>>(rr, pq, alphaB, flag);
    // x += a*p; r -= a*q
    k_axpy <<<EW(NX), 256, 0, stream>>>(x, r, p, q, alphaB, flag);
    // rr_new; beta; p = r + beta*p
    k_norm2<<<RED, 256, 0, stream>>>(r, partN, flag);
    k_fin1 <<<BN, RBLK, 0, stream>>>(partN, rrn, nullptr, flag);
    k_beta <<<1, 1, 0, stream>>>(rr, rrn, betaB, flag);
    k_pup  <<<EW(NX), 256, 0, stream>>>(p, r, betaB, flag);
  }
}